// MQA_20744692039964
// MI455X (gfx1250) — compile-verified
//
#include <hip/hip_runtime.h>
#include <math.h>

#define Bb 4
#define Tt 4096
#define Cc 1024
#define Hh 16
#define Dd 64
#define Ww 128
#define NBn (Tt / Ww)          // 32
#define NQKV (Cc + 2 * Dd)     // 1152

typedef __attribute__((ext_vector_type(2))) float v2f;
typedef __attribute__((ext_vector_type(8))) float v8f;
typedef int v4i __attribute__((vector_size(4 * sizeof(int))));

static __device__ __forceinline__ v8f wmma4(v2f a, v2f b, v8f c) {
  // D = A(16x4) * B(4x16) + C, fp32 exact
  return __builtin_amdgcn_wmma_f32_16x16x4_f32(false, a, false, b, (short)0, c,
                                               false, false);
}

// ---- CDNA5 async global->LDS staging (ASYNCcnt), with portable fallbacks ----
static __device__ __forceinline__ void async_copy16(const float* g, float* l) {
#if __has_builtin(__builtin_amdgcn_global_load_async_to_lds_b128)
  __builtin_amdgcn_global_load_async_to_lds_b128(
      (__attribute__((address_space(1))) v4i*)(v4i*)const_cast<float*>(g),
      (__attribute__((address_space(3))) v4i*)(v4i*)l, 0, 0);
#else
  *(float4*)l = *(const float4*)g;
#endif
}

static __device__ __forceinline__ void async_wait_le0() {
#if __has_builtin(__builtin_amdgcn_s_wait_asynccnt)
  __builtin_amdgcn_s_wait_asynccnt(0);
#elif __has_builtin(__builtin_amdgcn_global_load_async_to_lds_b128)
  asm volatile("s_wait_asynccnt 0x0" ::: "memory");
#endif
}

static __device__ __forceinline__ void async_wait_le2() {
#if __has_builtin(__builtin_amdgcn_s_wait_asynccnt)
  __builtin_amdgcn_s_wait_asynccnt(2);
#elif __has_builtin(__builtin_amdgcn_global_load_async_to_lds_b128)
  asm volatile("s_wait_asynccnt 0x2" ::: "memory");
#endif
}

// ---------------------------------------------------------------------------
// Generic row-major fp32 GEMM: C[M,N] = A[M,K] @ B[K,N]
// Block: 256 threads (8 waves), tile 64x64, K-step 16, async double-buffered.
// As: packed [64][16], XOR-swizzled on 4-float groups keyed on m[3:2] so the
// 16 lanes of an A-fragment b64 read hit 16 distinct banks.
// ---------------------------------------------------------------------------
__global__ __launch_bounds__(256) void gemm_f32_wmma(
    const float* __restrict__ A, const float* __restrict__ Bm,
    float* __restrict__ Cm, int M, int N, int K) {
  __shared__ float As[2][64 * 16];
  __shared__ float Bs[2][16 * 68];

  const int mtiles = M >> 6;
  const int bm = blockIdx.x % mtiles;
  const int bn = blockIdx.x / mtiles;
  const int row0 = bm * 64, col0 = bn * 64;

  const int tid = threadIdx.x;
  const int lane = tid & 31, wv = tid >> 5;
  const int la = lane & 15, lh = lane >> 4;
  const int mt = wv & 3;
  const int nt0 = (wv >> 2) * 2;

  const int arow = tid >> 2;                          // 0..63
  const int acol = (tid & 3) * 4;                     // 0..12
  const int asw = acol ^ (((arow >> 2) & 3) << 2);    // swizzled dest group
  const int brow = tid >> 4;                          // 0..15
  const int bcol = (tid & 15) * 4;                    // 0..60

  const int m = mt * 16 + la;
  const int msw = ((m >> 2) & 3) << 2;

  v8f acc0 = {0.f, 0.f, 0.f, 0.f, 0.f, 0.f, 0.f, 0.f};
  v8f acc1 = acc0;

  auto issue_stage = [&](int k0, int buf) {
    async_copy16(A + (size_t)(row0 + arow) * K + k0 + acol,
                 &As[buf][arow * 16 + asw]);
    async_copy16(Bm + (size_t)(k0 + brow) * N + col0 + bcol,
                 &Bs[buf][brow * 68 + bcol]);
  };

  const int KT = K >> 4;
  issue_stage(0, 0);
  for (int kt = 0; kt < KT; ++kt) {
    const int buf = kt & 1;
    __syncthreads();  // all waves done reading buf^1 -> safe to overwrite
    if (kt + 1 < KT) {
      issue_stage((kt + 1) * 16, buf ^ 1);
      async_wait_le2();  // current stage landed; next stage still in flight
    } else {
      async_wait_le0();
    }
    __syncthreads();  // current stage visible to all waves
#pragma unroll
    for (int ks = 0; ks < 4; ++ks) {
      const int koff = ks * 4 + lh * 2;
      v2f a, b0, b1;
      const float* as = &As[buf][m * 16];
      a.x = as[(koff ^ msw) + 0];
      a.y = as[(koff ^ msw) + 1];
      const float* bs = &Bs[buf][nt0 * 16 + la];
      b0.x = bs[(koff + 0) * 68];
      b0.y = bs[(koff + 1) * 68];
      b1.x = bs[(koff + 0) * 68 + 16];
      b1.y = bs[(koff + 1) * 68 + 16];
      acc0 = wmma4(a, b0, acc0);
      acc1 = wmma4(a, b1, acc1);
    }
  }
#pragma unroll
  for (int r = 0; r < 8; ++r) {
    const size_t mm = (size_t)(row0 + mt * 16 + lh * 8 + r);
    Cm[mm * N + col0 + nt0 * 16 + la] = acc0[r];
    Cm[mm * N + col0 + nt0 * 16 + 16 + la] = acc1[r];
  }
}

// ---------------------------------------------------------------------------
// In-place RMSNorm + RoPE on q (16 heads) and k slices of qkv.
// One wave32 per 64-wide row; lane holds elems d=lane and d=lane+32
// (exactly the rotate-half pairing used by RoPE).
// ---------------------------------------------------------------------------
__global__ __launch_bounds__(256) void rmsrope_qk(
    float* __restrict__ qkv, const int* __restrict__ pos_ids,
    const float* __restrict__ qw, const float* __restrict__ kw) {
  const int wid = blockIdx.x * 8 + (threadIdx.x >> 5);
  const int lane = threadIdx.x & 31;
  const int NQROWS = Bb * Tt * Hh;

  float* row;
  const float* w;
  int t;
  if (wid < NQROWS) {
    const int h = wid % Hh;
    const int bt = wid / Hh;
    t = bt % Tt;
    row = qkv + (size_t)bt * NQKV + h * Dd;
    w = qw;
  } else {
    const int bt = wid - NQROWS;
    if (bt >= Bb * Tt) return;
    t = bt % Tt;
    row = qkv + (size_t)bt * NQKV + Cc;
    w = kw;
  }

  float x1 = row[lane];
  float x2 = row[lane + 32];
  float s = x1 * x1 + x2 * x2;
#pragma unroll
  for (int mk = 16; mk >= 1; mk >>= 1) s += __shfl_xor(s, mk, 32);
  const float r = rsqrtf(s * (1.0f / 64.0f) + 1e-6f);
  x1 = x1 * r * w[lane];
  x2 = x2 * r * w[lane + 32];

  // inv_freq = 1e6^(-(2*lane)/64); axis = lane % 3
  const float invf = expf(-(float)(2 * lane) * (1.0f / 64.0f) * logf(1.0e6f));
  const float pos = (float)pos_ids[t * 3 + (lane % 3)];
  const float f = pos * invf;
  const float cs = cosf(f), sn = sinf(f);
  row[lane] = x1 * cs - x2 * sn;
  row[lane + 32] = x2 * cs + x1 * sn;
}

// ---------------------------------------------------------------------------
// Sliding-window block attention. One block per (b, h, n).
// Q: 128x64, KV: 256x64 (prev block + current block). 8 waves, each owns a
// 16-row q strip and the full 16x256 score strip in 16 v8f accumulators.
// Q/K/V staged via async global->LDS copies.
// ---------------------------------------------------------------------------
__global__ __launch_bounds__(256) void attn_mqa(const float* __restrict__ qkv,
                                                float* __restrict__ y) {
  extern __shared__ float sm[];
  float* sQ = sm;                 // 128*68
  float* sK = sQ + 128 * 68;      // 256*68
  float* sV = sK + 256 * 68;      // 256*68
  float* sP = sV + 256 * 68;      // 8 * 16*17 per-wave transpose scratch

  const int blk = blockIdx.x;
  const int n = blk % NBn;
  const int h = (blk / NBn) % Hh;
  const int b = blk / (NBn * Hh);

  const int tid = threadIdx.x, lane = tid & 31, wv = tid >> 5;
  const int la = lane & 15, lh = lane >> 4;

  // ---- stage Q, K, V into LDS (async DMA path) ----
  {
    const int col4 = (tid & 15) * 4;
    const int rbase = tid >> 4;  // 0..15
#pragma unroll
    for (int i = 0; i < 8; ++i) {
      const int rr = rbase + 16 * i;  // 0..127
      async_copy16(
          qkv + (size_t)(b * Tt + n * Ww + rr) * NQKV + h * Dd + col4,
          sQ + rr * 68 + col4);
    }
#pragma unroll
    for (int i = 0; i < 16; ++i) {
      const int rr = rbase + 16 * i;  // 0..255
      const int tg = (n - 1) * Ww + rr;
      if (tg >= 0) {
        const size_t base = (size_t)(b * Tt + tg) * NQKV;
        async_copy16(qkv + base + Cc + col4, sK + rr * 68 + col4);
        async_copy16(qkv + base + Cc + Dd + col4, sV + rr * 68 + col4);
      } else {
        const float4 z = make_float4(0.f, 0.f, 0.f, 0.f);
        *(float4*)(sK + rr * 68 + col4) = z;
        *(float4*)(sV + rr * 68 + col4) = z;
      }
    }
    async_wait_le0();
  }
  __syncthreads();

  // ---- scores: S = Q @ K^T  (wave strip 16x256, 16 accumulator tiles) ----
  const int qrow0 = wv * 16;
  v8f acc[16];
  const v8f vzero = {0.f, 0.f, 0.f, 0.f, 0.f, 0.f, 0.f, 0.f};
#pragma unroll
  for (int t = 0; t < 16; ++t) acc[t] = vzero;

  for (int d0 = 0; d0 < 64; d0 += 4) {
    v2f a;
    a.x = sQ[(qrow0 + la) * 68 + d0 + lh * 2];
    a.y = sQ[(qrow0 + la) * 68 + d0 + lh * 2 + 1];
#pragma unroll
    for (int t = 0; t < 16; ++t) {
      v2f bf;  // B[d][kr] = K[kr][d]
      bf.x = sK[(t * 16 + la) * 68 + d0 + lh * 2];
      bf.y = sK[(t * 16 + la) * 68 + d0 + lh * 2 + 1];
      acc[t] = wmma4(a, bf, acc[t]);
    }
  }

  // ---- mask + softmax (rows live on 16-lane halves: shfl_xor 1,2,4,8) ----
  const float scale = 0.125f;  // 1/sqrt(64)
  float rowmax[8], rowsum[8];
#pragma unroll
  for (int r = 0; r < 8; ++r) rowmax[r] = -3.0e38f;
#pragma unroll
  for (int t = 0; t < 16; ++t) {
    const int kr = t * 16 + la;
#pragma unroll
    for (int r = 0; r < 8; ++r) {
      const int qr = qrow0 + lh * 8 + r;
      const bool ok = (kr >= qr) && (kr <= qr + Ww) && ((n > 0) || (kr >= Ww));
      const float sv = ok ? acc[t][r] * scale : -1.0e30f;
      acc[t][r] = sv;
      rowmax[r] = fmaxf(rowmax[r], sv);
    }
  }
#pragma unroll
  for (int r = 0; r < 8; ++r) {
#pragma unroll
    for (int mk = 1; mk <= 8; mk <<= 1)
      rowmax[r] = fmaxf(rowmax[r], __shfl_xor(rowmax[r], mk, 32));
    rowsum[r] = 0.f;
  }
#pragma unroll
  for (int t = 0; t < 16; ++t) {
#pragma unroll
    for (int r = 0; r < 8; ++r) {
      const float p = __expf(acc[t][r] - rowmax[r]);
      acc[t][r] = p;
      rowsum[r] += p;
    }
  }
#pragma unroll
  for (int r = 0; r < 8; ++r) {
#pragma unroll
    for (int mk = 1; mk <= 8; mk <<= 1)
      rowsum[r] += __shfl_xor(rowsum[r], mk, 32);
    rowsum[r] = 1.0f / rowsum[r];
  }
#pragma unroll
  for (int t = 0; t < 16; ++t) {
#pragma unroll
    for (int r = 0; r < 8; ++r) acc[t][r] *= rowsum[r];
  }

  // ---- P @ V : bounce each 16x16 P tile through per-wave LDS (D->A relayout)
  float* sPw = sP + wv * (16 * 17);
  v8f yacc[4];
#pragma unroll
  for (int nt = 0; nt < 4; ++nt) yacc[nt] = vzero;

  for (int t = 0; t < 16; ++t) {
#pragma unroll
    for (int r = 0; r < 8; ++r) sPw[(lh * 8 + r) * 17 + la] = acc[t][r];
    asm volatile("s_wait_dscnt 0x0" ::: "memory");
#pragma unroll
    for (int ks = 0; ks < 4; ++ks) {
      v2f a;
      a.x = sPw[la * 17 + ks * 4 + lh * 2];
      a.y = sPw[la * 17 + ks * 4 + lh * 2 + 1];
#pragma unroll
      for (int nt = 0; nt < 4; ++nt) {
        v2f bf;  // B[k][d] = V[16t + klocal][d]
        bf.x = sV[(t * 16 + ks * 4 + lh * 2 + 0) * 68 + nt * 16 + la];
        bf.y = sV[(t * 16 + ks * 4 + lh * 2 + 1) * 68 + nt * 16 + la];
        yacc[nt] = wmma4(a, bf, yacc[nt]);
      }
    }
    asm volatile("s_wait_dscnt 0x0" ::: "memory");
  }

  // ---- store y in (B, T, H*D) layout ----
#pragma unroll
  for (int nt = 0; nt < 4; ++nt) {
#pragma unroll
    for (int r = 0; r < 8; ++r) {
      const int qr = qrow0 + lh * 8 + r;
      y[(size_t)(b * Tt + n * Ww + qr) * Cc + h * Dd + nt * 16 + la] =
          yacc[nt][r];
    }
  }
}

// ---------------------------------------------------------------------------
extern "C" void kernel_launch(void* const* d_in, const int* in_sizes, int n_in,
                              void* d_out, int out_size, void* d_ws,
                              size_t ws_size, hipStream_t stream) {
  const float* x = (const float*)d_in[0];
  const int* pos_ids = (const int*)d_in[1];
  const float* w_attn = (const float*)d_in[2];
  const float* w_proj = (const float*)d_in[3];
  const float* qnw = (const float*)d_in[4];
  const float* knw = (const float*)d_in[5];
  float* out = (float*)d_out;

  float* qkv = (float*)d_ws;                      // B*T*1152 floats
  float* yatt = qkv + (size_t)Bb * Tt * NQKV;     // B*T*1024 floats

  const int M = Bb * Tt;  // 16384

  // 1) qkv = x @ w_attn   (16384 x 1024 x 1152)
  gemm_f32_wmma<<<dim3((M / 64) * (NQKV / 64)), dim3(256), 0, stream>>>(
      x, w_attn, qkv, M, NQKV, Cc);

  // 2) in-place RMSNorm + RoPE on q heads and k
  {
    const int rows = Bb * Tt * (Hh + 1);
    rmsrope_qk<<<dim3((rows + 7) / 8), dim3(256), 0, stream>>>(qkv, pos_ids,
                                                               qnw, knw);
  }

  // 3) block attention
  {
    const size_t shmem =
        (size_t)(128 * 68 + 256 * 68 + 256 * 68 + 8 * 16 * 17) * sizeof(float);
    (void)hipFuncSetAttribute((const void*)attn_mqa,
                              hipFuncAttributeMaxDynamicSharedMemorySize,
                              (int)shmem);
    attn_mqa<<<dim3(Bb * Hh * NBn), dim3(256), shmem, stream>>>(qkv, yatt);
  }

  // 4) out = yatt @ w_proj  (16384 x 1024 x 1024)
  gemm_f32_wmma<<<dim3((M / 64) * (Cc / 64)), dim3(256), 0, stream>>>(
      yatt, w_proj, out, M, Cc, Cc);
}